// NStepReturn_88691074662796
// MI455X (gfx1250) — compile-verified
//
#include <hip/hip_runtime.h>
#include <hip/hip_bf16.h>

// ---------------------------------------------------------------------------
// n-step discounted return as a banded matmul on V_WMMA_F32_16X16X4_F32.
//
//   G[t, b] = sum_{h=0..15} g^h * rm[t+h, b]  +  g^eff(t) * V[min(t+15,T-1), b]
//
// For a 16x16 (time x batch) tile:  G = L0*RM_tile0 + L1*RM_tile1
//   L0[i,k] = g^(k-i) for k>=i else 0
//   L1[i,j] = g^(16+j-i) for j<i else 0
// Two 16x16x16 f32 matmuls = 8 v_wmma_f32_16x16x4_f32 per output tile.
// Bandwidth-bound (~68 MB moved -> ~2.9 us at 23.3 TB/s); f32 WMMA keeps
// full reference precision. All gamma powers are branchless bit-product
// constants -- no transcendentals anywhere.
// ---------------------------------------------------------------------------

typedef __attribute__((ext_vector_type(2))) float v2f;
typedef __attribute__((ext_vector_type(8))) float v8f;

constexpr int   T_DIM    = 1024;
constexpr int   B_DIM    = 4096;
constexpr int   HORIZON  = 16;
constexpr int   TT_TILES = T_DIM / 16;       // 64 time tiles
constexpr int   CT_TILES = B_DIM / 16;       // 256 batch-column tiles
constexpr int   CHUNK    = 8;                // time tiles per wave
constexpr int   NCHUNK   = TT_TILES / CHUNK; // 8 chunks -> 2048 waves total

constexpr float G1  = 0.99f;
constexpr float G2  = 0.9801f;               // 0.99^2
constexpr float G4  = 0.96059601f;           // 0.99^4
constexpr float G8  = 0.9227446944279201f;   // 0.99^8
constexpr float G16 = 0.8514577710948755f;   // 0.99^16

// gamma^e for e in [0,15]: branchless product over set bits (7 VALU, no trans)
__device__ __forceinline__ float gpow4(int e) {
  float p = (e & 1) ? G1 : 1.0f;
  p *= (e & 2) ? G2 : 1.0f;
  p *= (e & 4) ? G4 : 1.0f;
  p *= (e & 8) ? G8 : 1.0f;
  return p;
}

// Load one 16x16 rm tile (rewards*masks) laid out as four 4x16 B-operands.
// B operand layout: lane l, reg r  ->  B[2*(l>=16) + r, l&15]   (K, N)
__device__ __forceinline__ void load_rm_tile(const float* __restrict__ rewards,
                                             const float* __restrict__ masks,
                                             int t0, int col, int half,
                                             v2f b[4]) {
  if (t0 >= T_DIM) {               // window runs past T: padded zeros
#pragma unroll
    for (int q = 0; q < 4; ++q) { b[q][0] = 0.0f; b[q][1] = 0.0f; }
    return;
  }
#pragma unroll
  for (int q = 0; q < 4; ++q) {
#pragma unroll
    for (int r = 0; r < 2; ++r) {
      const int row = t0 + 4 * q + 2 * half + r;   // K within the 16x16 tile
      const int idx = row * B_DIM + col;
      b[q][r] = rewards[idx] * masks[idx];
    }
  }
}

__global__ __launch_bounds__(256) void nstep_return_wmma(
    const float* __restrict__ rewards,
    const float* __restrict__ values,
    const float* __restrict__ masks,
    float* __restrict__ out) {
  const int tid  = blockIdx.x * blockDim.x + threadIdx.x;
  const int wave = tid >> 5;
  const int lane = tid & 31;
  const int half = lane >> 4;   // lane half selects K pairs / M+8
  const int li   = lane & 15;

  const int col_tile = wave / NCHUNK;       // 0..255
  const int chunk    = wave % NCHUNK;       // 0..7
  const int col      = col_tile * 16 + li;  // this lane's batch column
  const int tt_begin = chunk * CHUNK;

  // --- Discount coefficient matrices as A-operands (per-lane constants). ---
  // A operand layout: lane l, reg r -> A[l&15, 2*(l>=16)+r]; chunk q adds 4q.
  // For d = K - M: L0 needs gamma^d (d>=0), L1 needs gamma^(16+d) (d<0);
  // both exponents equal (d & 15), so one gpow4 feeds both matrices.
  v2f aL0[4], aL1[4];
#pragma unroll
  for (int q = 0; q < 4; ++q) {
#pragma unroll
    for (int r = 0; r < 2; ++r) {
      const int Kg = 4 * q + 2 * half + r;
      const int d  = Kg - li;
      const float p = gpow4(d & 15);
      aL0[q][r] = (d >= 0) ? p : 0.0f;
      aL1[q][r] = (d < 0)  ? p : 0.0f;
    }
  }

  // Rolling rm tiles: tile tt is reused as tile1 of tt-1 and tile0 of tt.
  v2f cur[4], nxt[4];
  load_rm_tile(rewards, masks, tt_begin * 16, col, half, cur);

  for (int tt = tt_begin; tt < tt_begin + CHUNK; ++tt) {
    const int t0 = tt * 16;

    // Streaming hint for the tile after next (global_prefetch_b8).
    const int pf_row = (tt + 2) * 16 + 2 * half;
    if (pf_row < T_DIM) {
      __builtin_prefetch(&rewards[pf_row * B_DIM + col], 0, 0);
      __builtin_prefetch(&masks[pf_row * B_DIM + col], 0, 0);
    }

    load_rm_tile(rewards, masks, (tt + 1) * 16, col, half, nxt);

    // G = L0*cur + L1*nxt : 8x v_wmma_f32_16x16x4_f32, f32 accumulate.
    v8f acc = {};
#pragma unroll
    for (int q = 0; q < 4; ++q)
      acc = __builtin_amdgcn_wmma_f32_16x16x4_f32(
          false, aL0[q], false, cur[q], (short)0, acc, false, false);
#pragma unroll
    for (int q = 0; q < 4; ++q)
      acc = __builtin_amdgcn_wmma_f32_16x16x4_f32(
          false, aL1[q], false, nxt[q], (short)0, acc, false, false);

    // Bootstrap + store. C/D layout: reg r, lane l -> M = r + 8*half, N = li.
    if (tt < TT_TILES - 1) {
      // All rows: eff = 16 (constant gamma^16), vidx = t + 15 <= T-2.
#pragma unroll
      for (int r = 0; r < 8; ++r) {
        const int t = t0 + r + 8 * half;
        out[t * B_DIM + col] =
            acc[r] + G16 * values[(t + HORIZON - 1) * B_DIM + col];
      }
    } else {
      // Final time tile: t = 1008+m; vidx clamps to T-1; eff = min(16, T-t).
      const float vlast = values[(T_DIM - 1) * B_DIM + col];
#pragma unroll
      for (int r = 0; r < 8; ++r) {
        const int m = r + 8 * half;
        const int t = t0 + m;
        const float coef = (m == 0) ? G16 : gpow4(HORIZON - m);
        out[t * B_DIM + col] = acc[r] + coef * vlast;
      }
    }

#pragma unroll
    for (int q = 0; q < 4; ++q) cur[q] = nxt[q];
  }
}

extern "C" void kernel_launch(void* const* d_in, const int* in_sizes, int n_in,
                              void* d_out, int out_size, void* d_ws,
                              size_t ws_size, hipStream_t stream) {
  (void)in_sizes; (void)n_in; (void)out_size; (void)d_ws; (void)ws_size;
  const float* rewards = (const float*)d_in[0];
  const float* values  = (const float*)d_in[1];
  const float* masks   = (const float*)d_in[2];
  float* out = (float*)d_out;

  const int total_waves = CT_TILES * NCHUNK;          // 2048 waves
  dim3 block(256);                                    // 8 wave32 per block
  dim3 grid((total_waves * 32) / 256);                // 256 blocks
  nstep_return_wmma<<<grid, block, 0, stream>>>(rewards, values, masks, out);
}